// QuanvolutionFilter_65481071408558
// MI455X (gfx1250) — compile-verified
//
#include <hip/hip_runtime.h>

// Quanvolution closed-form: per 2x2 patch p, out = cumprod(cos(theta * p)).
// Memory-bound: ~205 MB traffic -> ~8.8 us floor @ 23.3 TB/s. cos runs on the
// quarter-rate TRANS pipe (103M ops), so per-patch VALU overhead is trimmed to
// nothing: all loop addressing is base + compile-time immediate.
//
// Main kernel: 224 threads (7 wave32s) stage 8 images (25088 B) into LDS via
// global_load_async_to_lds_b128 (single base VGPR pair, offset:i*3584 applies
// to both global and LDS per CDNA5 ISA), s_wait_asynccnt 0 +
// barrier, then 7 patches/thread from LDS with ds_load_b64 pairs and
// non-temporal global_store_b128.

#define INV_2PI 0.15915494309189535f
#define IMGS_PER_BLOCK 8
#define THREADS 224                      // 7 wave32s; 1568 chunks/patches = 7/thread
#define TILE_FLOATS (IMGS_PER_BLOCK * 784)

typedef float v4f __attribute__((ext_vector_type(4)));
typedef float v2f __attribute__((ext_vector_type(2)));

// Stringified-immediate async copy: both LDS and global address get +OFF.
#define ASYNC_B128(lds, gaddr, OFF)                                   \
  asm volatile("global_load_async_to_lds_b128 %0, %1, off offset:" #OFF \
               :                                                      \
               : "v"(lds), "v"(gaddr)                                 \
               : "memory")

__global__ __launch_bounds__(THREADS) void quanv_main(
    const float* __restrict__ x, const float* __restrict__ theta,
    float* __restrict__ out) {
  __shared__ float tile[TILE_FLOATS];  // 25088 bytes

  const int tid = threadIdx.x;

  // ---- Async stage: 1568 x 16B chunks, 7 per thread, immediate offsets ----
  const unsigned lds0 =
      (unsigned)(size_t)(&tile[0]) + (unsigned)tid * 16u;
  const unsigned long long g0 =
      (unsigned long long)(size_t)(x + (size_t)blockIdx.x * TILE_FLOATS) +
      (unsigned long long)tid * 16ull;
  ASYNC_B128(lds0, g0, 0);
  ASYNC_B128(lds0, g0, 3584);
  ASYNC_B128(lds0, g0, 7168);
  ASYNC_B128(lds0, g0, 10752);
  ASYNC_B128(lds0, g0, 14336);
  ASYNC_B128(lds0, g0, 17920);
  ASYNC_B128(lds0, g0, 21504);
  asm volatile("s_wait_asynccnt 0" ::: "memory");
  __syncthreads();

  // theta is uniform -> s_load; fold v_cos_f32's 1/(2*pi) revolution scale in.
  const float s0 = theta[0] * INV_2PI;
  const float s1 = theta[1] * INV_2PI;
  const float s2 = theta[2] * INV_2PI;
  const float s3 = theta[3] * INV_2PI;

  // Patch p = tid + i*224. LDS float addr = 4p - 2c with c = p % 14 invariant
  // across i (224 % 14 == 0), so the loop strides by exactly 896 floats.
  const int u = (tid >= 196) ? tid - 196 : tid;
  const int c = u % 14;
  const float* tb = tile + 4 * tid - 2 * c;
  v4f* dst = (v4f*)out + (size_t)blockIdx.x * (IMGS_PER_BLOCK * 196) + tid;

#pragma unroll
  for (int i = 0; i < 7; ++i) {
    const v2f top = *(const v2f*)(tb + i * 896);       // ds_load_b64
    const v2f bot = *(const v2f*)(tb + i * 896 + 28);  // ds_load_b64
    const float c0 = __builtin_amdgcn_cosf(top.x * s0);  // v_cos_f32
    const float c1 = __builtin_amdgcn_cosf(top.y * s1);
    const float c2 = __builtin_amdgcn_cosf(bot.x * s2);
    const float c3 = __builtin_amdgcn_cosf(bot.y * s3);
    v4f o;
    o.x = c0;
    o.y = c0 * c1;
    o.z = o.y * c2;
    o.w = o.z * c3;
    __builtin_nontemporal_store(o, dst + i * 224);  // b128 th:NT, no reuse
  }
}

// Generic guarded tail for batch sizes not divisible by IMGS_PER_BLOCK.
__global__ void quanv_tail(const float* __restrict__ x,
                           const float* __restrict__ theta,
                           float* __restrict__ out, int img_start, int nimg) {
  const long long idx = (long long)blockIdx.x * blockDim.x + threadIdx.x;
  const long long npatch = (long long)(nimg - img_start) * 196;
  if (idx >= npatch) return;
  const long long p = (long long)img_start * 196 + idx;
  const long long img = p / 196;
  const int u = (int)(p - img * 196);
  const int r = u / 14;
  const int cc = u - r * 14;
  const float* b = x + img * 784 + (2 * r) * 28 + 2 * cc;
  const v2f top = *(const v2f*)(b);
  const v2f bot = *(const v2f*)(b + 28);
  const float c0 = __builtin_amdgcn_cosf(top.x * (theta[0] * INV_2PI));
  const float c1 = __builtin_amdgcn_cosf(top.y * (theta[1] * INV_2PI));
  const float c2 = __builtin_amdgcn_cosf(bot.x * (theta[2] * INV_2PI));
  const float c3 = __builtin_amdgcn_cosf(bot.y * (theta[3] * INV_2PI));
  v4f o;
  o.x = c0;
  o.y = c0 * c1;
  o.z = o.y * c2;
  o.w = o.z * c3;
  ((v4f*)out)[p] = o;
}

extern "C" void kernel_launch(void* const* d_in, const int* in_sizes, int n_in,
                              void* d_out, int out_size, void* d_ws, size_t ws_size,
                              hipStream_t stream) {
  const float* x = (const float*)d_in[0];      // (B, 28, 28) f32
  const float* theta = (const float*)d_in[1];  // (4,) f32
  float* out = (float*)d_out;                  // (B, 784) f32
  const int nimg = in_sizes[0] / 784;

  const int full_blocks = nimg / IMGS_PER_BLOCK;  // 4096 for B=32768
  if (full_blocks > 0) {
    quanv_main<<<full_blocks, THREADS, 0, stream>>>(x, theta, out);
  }
  const int img_start = full_blocks * IMGS_PER_BLOCK;
  if (img_start < nimg) {
    const long long npatch = (long long)(nimg - img_start) * 196;
    const int tb = 256;
    const int tg = (int)((npatch + tb - 1) / tb);
    quanv_tail<<<tg, tb, 0, stream>>>(x, theta, out, img_start, nimg);
  }
}